// ssGCN_63797444215684
// MI455X (gfx1250) — compile-verified
//
#include <hip/hip_runtime.h>
#include <hip/hip_bf16.h>
#include <cstdint>

typedef __attribute__((ext_vector_type(2))) float v2f;
typedef __attribute__((ext_vector_type(8))) float v8f;

#define WMMA_F32(a, b, c) \
  __builtin_amdgcn_wmma_f32_16x16x4_f32(false, (a), false, (b), (short)0, (c), false, false)

// ---- WMMA fragment helpers (fp32 16x16x4, wave32) -------------------------
// A 16x4 (MxK): lane L holds A[L&15][2*(L>>4) + {0,1}]  (ISA 7.12.2)
__device__ __forceinline__ v2f lda_frag(const float* t, int lane, int ks, int ld) {
  int m = lane & 15;
  int k = ks * 4 + 2 * (lane >> 4);
  v2f a;
  a.x = t[m * ld + k];
  a.y = t[m * ld + k + 1];
  return a;
}
// B 4x16 (KxN): lane L holds B[2*(L>>4) + {0,1}][L&15]
__device__ __forceinline__ v2f ldb_frag(const float* w, int lane, int ks, int n0, int ld) {
  int n = n0 + (lane & 15);
  int k = ks * 4 + 2 * (lane >> 4);
  v2f b;
  b.x = w[k * ld + n];
  b.y = w[(k + 1) * ld + n];
  return b;
}
// cooperative contiguous copy (16B granules) -- block-shared weights
__device__ __forceinline__ void copy_f4(float* dst, const float* src, int nfl, int tid, int nthr) {
  float4* d = (float4*)dst;
  const float4* s = (const float4*)src;
  for (int i = tid; i < nfl / 4; i += nthr) d[i] = s[i];
}

// ---- CDNA5 async global->LDS tile staging (ASYNCcnt path) -----------------
// Stage ROWS x RF contiguous floats from global into an LDS tile padded to LD
// floats per row. Rows are 16B aligned (LD % 4 == 0), so B128 is legal.
template <int ROWS, int RF, int LD>
__device__ __forceinline__ void async_stage_tile(float* dst, const float* gsrc, int lane) {
  constexpr int CH = ROWS * (RF / 4);  // float4 chunks
  for (int i = lane; i < CH; i += 32) {
    int r = i / (RF / 4);
    int c = i % (RF / 4);
    unsigned ldsa = (unsigned)(uintptr_t)(dst + r * LD + c * 4);  // low 32 bits = LDS addr
    asm volatile("global_load_async_to_lds_b128 %0, %1, off"
                 :: "v"(ldsa), "v"(gsrc + i * 4)
                 : "memory");
  }
}
__device__ __forceinline__ void wait_async_lds() {
  asm volatile("s_wait_asynccnt 0x0" ::: "memory");
}

// Padded leading dims: (m*LD + k) % 64 distinct across the 16 A-frag lanes.
#define LD64 68   // 16x64 tiles
#define LD32 36   // 16x32 tiles
#define LD16 18   // 16x16 tiles

// ---------------------------------------------------------------------------
// SE layer: xu = x@Uw + Ub ; w = sigmoid(relu(xu@f1 + b1)@f2 + b2) ; h = xu*w
// 128 threads = 4 waves, 16 nodes per wave.
__global__ __launch_bounds__(128) void se_kernel(
    const float* __restrict__ x, const float* __restrict__ Uw, const float* __restrict__ Ub,
    const float* __restrict__ f1w, const float* __restrict__ f1b,
    const float* __restrict__ f2w, const float* __restrict__ f2b,
    float* __restrict__ h_out, float* __restrict__ w_out, int N) {
  __shared__ __align__(16) float sUw[64 * 64];
  __shared__ __align__(16) float sF1[64 * 16];
  __shared__ __align__(16) float sF2[16 * 64];
  __shared__ __align__(16) float sXU[4][16 * LD64];
  __shared__ __align__(16) float sT[4][16 * LD16];

  const int tid = threadIdx.x, nthr = blockDim.x;
  copy_f4(sUw, Uw, 64 * 64, tid, nthr);
  copy_f4(sF1, f1w, 64 * 16, tid, nthr);
  copy_f4(sF2, f2w, 16 * 64, tid, nthr);
  __syncthreads();

  const int wid = tid >> 5, lane = tid & 31;
  const int col = lane & 15, rlo = (lane >> 4) * 8;
  const long row0 = (long)blockIdx.x * 64 + wid * 16;
  const bool active = row0 < N;        // N % 16 == 0 -> tiles fully in or out
  const long r0 = active ? row0 : 0;   // clamp so OOB waves still do valid loads

  float* xt = sXU[wid];
  async_stage_tile<16, 64, LD64>(xt, x + r0 * 64, lane);  // tile is contiguous in global
  wait_async_lds();

  // A fragments of the 16x64 x-tile (whole tile lives in wave registers)
  v2f xa[16];
  for (int ks = 0; ks < 16; ++ks) xa[ks] = lda_frag(xt, lane, ks, LD64);

  // xu = x @ Uw + Ub   (4 n-tiles, K=64)
  v8f xu[4];
  for (int nt = 0; nt < 4; ++nt) {
    v8f c = {};
    for (int ks = 0; ks < 16; ++ks)
      c = WMMA_F32(xa[ks], ldb_frag(sUw, lane, ks, nt * 16, 64), c);
    float bias = Ub[nt * 16 + col];
    for (int j = 0; j < 8; ++j) c[j] += bias;
    xu[nt] = c;
  }
  // park xu in LDS (overwrites x-tile; x fully consumed via xa[])
  for (int nt = 0; nt < 4; ++nt)
    for (int j = 0; j < 8; ++j) xt[(j + rlo) * LD64 + nt * 16 + col] = xu[nt][j];

  // t = relu(xu @ f1 + b1)   16x16, K=64
  v8f t = {};
  for (int ks = 0; ks < 16; ++ks)
    t = WMMA_F32(lda_frag(xt, lane, ks, LD64), ldb_frag(sF1, lane, ks, 0, 16), t);
  {
    float bias = f1b[col];
    for (int j = 0; j < 8; ++j) t[j] = fmaxf(t[j] + bias, 0.0f);
  }
  float* tt = sT[wid];
  for (int j = 0; j < 8; ++j) tt[(j + rlo) * LD16 + col] = t[j];

  // w = sigmoid(t @ f2 + b2)   K=16
  v8f wt[4];
  for (int nt = 0; nt < 4; ++nt) {
    v8f c = {};
    for (int ks = 0; ks < 4; ++ks)
      c = WMMA_F32(lda_frag(tt, lane, ks, LD16), ldb_frag(sF2, lane, ks, nt * 16, 64), c);
    float bias = f2b[nt * 16 + col];
    for (int j = 0; j < 8; ++j) c[j] = 1.0f / (1.0f + __expf(-(c[j] + bias)));
    wt[nt] = c;
  }

  if (active) {
    for (int nt = 0; nt < 4; ++nt)
      for (int j = 0; j < 8; ++j) {
        long row = row0 + j + rlo;
        int cc = nt * 16 + col;
        float wv = wt[nt][j];
        w_out[row * 64 + cc] = wv;
        h_out[row * 64 + cc] = xu[nt][j] * wv;
      }
  }
}

// ---------------------------------------------------------------------------
// C[N x NO] = A[N x 64] @ W[64 x NO]   (bias handled later, per reference)
template <int NO>
__global__ __launch_bounds__(256) void gemm_kernel(const float* __restrict__ A,
                                                   const float* __restrict__ W,
                                                   float* __restrict__ out, int N) {
  __shared__ __align__(16) float sW[64 * NO];
  __shared__ __align__(16) float sA[8][16 * LD64];
  const int tid = threadIdx.x;
  copy_f4(sW, W, 64 * NO, tid, blockDim.x);
  __syncthreads();

  const int wid = tid >> 5, lane = tid & 31;
  const int col = lane & 15, rlo = (lane >> 4) * 8;
  const long row0 = (long)blockIdx.x * 128 + wid * 16;
  const bool active = row0 < N;
  const long r0 = active ? row0 : 0;

  float* at = sA[wid];
  async_stage_tile<16, 64, LD64>(at, A + r0 * 64, lane);
  wait_async_lds();

  v2f af[16];
  for (int ks = 0; ks < 16; ++ks) af[ks] = lda_frag(at, lane, ks, LD64);

  for (int nt = 0; nt < NO / 16; ++nt) {
    v8f c = {};
    for (int ks = 0; ks < 16; ++ks)
      c = WMMA_F32(af[ks], ldb_frag(sW, lane, ks, nt * 16, NO), c);
    if (active) {
      for (int j = 0; j < 8; ++j)
        out[(row0 + j + rlo) * NO + nt * 16 + col] = c[j];
    }
  }
}

// ---------------------------------------------------------------------------
__global__ void fill_zero(float* p, long n) {
  long i = (long)blockIdx.x * blockDim.x + threadIdx.x;
  if (i < n) p[i] = 0.0f;
}

__global__ void degree_kernel(const long long* __restrict__ ei, float* deg, int E) {
  int e = blockIdx.x * blockDim.x + threadIdx.x;
  if (e < E) atomicAdd(&deg[(long)ei[E + e]], 1.0f);
}

__global__ void dinv_kernel(const float* __restrict__ deg, float* dinv, int N) {
  int i = blockIdx.x * blockDim.x + threadIdx.x;
  if (i < N) dinv[i] = rsqrtf(deg[i] + 1.0f);  // +1 self-loop; always > 0
}

// agg[dst] += hw[src] * dinv[src]*dinv[dst]  (NO/4 threads per edge, float4 gather)
template <int NO>
__global__ void aggregate_kernel(const long long* __restrict__ ei,
                                 const float* __restrict__ hw,
                                 const float* __restrict__ dinv,
                                 float* __restrict__ agg, int E) {
  const int TPE = NO / 4;
  long tid = (long)blockIdx.x * blockDim.x + threadIdx.x;
  long e = tid / TPE;
  int t = (int)(tid % TPE);
  if (e >= E) return;
  long s = (long)ei[e];
  long d = (long)ei[E + e];
  float coef = dinv[s] * dinv[d];
  float4 v = ((const float4*)(hw + s * NO))[t];
  float* base = agg + d * NO + t * 4;
  atomicAdd(base + 0, v.x * coef);
  atomicAdd(base + 1, v.y * coef);
  atomicAdd(base + 2, v.z * coef);
  atomicAdd(base + 3, v.w * coef);
}

// out = relu(agg + hw*dinv^2 + b)
__global__ void finalize_kernel(const float* __restrict__ agg, const float* __restrict__ hw,
                                const float* __restrict__ dinv, const float* __restrict__ bias,
                                float* __restrict__ out, long n, int NO) {
  long i = (long)blockIdx.x * blockDim.x + threadIdx.x;
  if (i >= n) return;
  long r = i / NO;
  int c = (int)(i % NO);
  float di = dinv[r];
  out[i] = fmaxf(agg[i] + hw[i] * di * di + bias[c], 0.0f);
}

// ---------------------------------------------------------------------------
// logits = h2 @ fcw + fcb ; log_softmax over 16 cols (cols live across 16 lanes)
__global__ __launch_bounds__(256) void classifier_kernel(const float* __restrict__ h2,
                                                         const float* __restrict__ fcw,
                                                         const float* __restrict__ fcb,
                                                         float* __restrict__ out, int N) {
  __shared__ __align__(16) float sW[32 * 16];
  __shared__ __align__(16) float sA[8][16 * LD32];
  const int tid = threadIdx.x;
  copy_f4(sW, fcw, 32 * 16, tid, blockDim.x);
  __syncthreads();

  const int wid = tid >> 5, lane = tid & 31;
  const int col = lane & 15, rlo = (lane >> 4) * 8;
  const long row0 = (long)blockIdx.x * 128 + wid * 16;
  const bool active = row0 < N;
  const long r0 = active ? row0 : 0;

  float* at = sA[wid];
  async_stage_tile<16, 32, LD32>(at, h2 + r0 * 32, lane);
  wait_async_lds();

  v8f c = {};
  for (int ks = 0; ks < 8; ++ks)
    c = WMMA_F32(lda_frag(at, lane, ks, LD32), ldb_frag(sW, lane, ks, 0, 16), c);
  {
    float bias = fcb[col];
    for (int j = 0; j < 8; ++j) c[j] += bias;
  }
  // log_softmax: each output row's 16 values sit in one 16-lane half at fixed j
  for (int j = 0; j < 8; ++j) {
    float m = c[j];
    for (int msk = 8; msk; msk >>= 1) m = fmaxf(m, __shfl_xor(m, msk, 32));
    float ex = __expf(c[j] - m);
    float ssum = ex;
    for (int msk = 8; msk; msk >>= 1) ssum += __shfl_xor(ssum, msk, 32);
    c[j] = (c[j] - m) - __logf(ssum);
  }
  if (active) {
    for (int j = 0; j < 8; ++j) out[(row0 + j + rlo) * 16 + col] = c[j];
  }
}

// ---------------------------------------------------------------------------
extern "C" void kernel_launch(void* const* d_in, const int* in_sizes, int n_in,
                              void* d_out, int out_size, void* d_ws, size_t ws_size,
                              hipStream_t stream) {
  const float* x = (const float*)d_in[0];
  const long long* ei = (const long long*)d_in[1];  // int64 [2, E]
  const float* Uw = (const float*)d_in[2];
  const float* Ub = (const float*)d_in[3];
  const float* f1w = (const float*)d_in[4];
  const float* f1b = (const float*)d_in[5];
  const float* f2w = (const float*)d_in[6];
  const float* f2b = (const float*)d_in[7];
  const float* W1 = (const float*)d_in[8];
  const float* b1 = (const float*)d_in[9];
  const float* W2 = (const float*)d_in[10];
  const float* b2 = (const float*)d_in[11];
  const float* fcw = (const float*)d_in[12];
  const float* fcb = (const float*)d_in[13];

  const int N = in_sizes[0] / 64;
  const int E = in_sizes[1] / 2;

  float* out = (float*)d_out;
  float* out_logits = out;                // [N,16]
  float* out_w = out + (long)N * 16;      // [N,64]

  // workspace layout (floats): A=64N, B=64N, C=64N, deg=N, dinv=N  (~78 MB)
  float* bufA = (float*)d_ws;                 // h, later h1
  float* bufB = bufA + (long)64 * N;          // hw1; later hw2[0:32N], h2[32N:64N]
  float* bufC = bufB + (long)64 * N;          // agg1; later agg2[0:32N]
  float* deg = bufC + (long)64 * N;
  float* dinv = deg + N;

  // --- SE layer: bufA = h, out_w = w
  se_kernel<<<(N + 63) / 64, 128, 0, stream>>>(x, Uw, Ub, f1w, f1b, f2w, f2b, bufA, out_w, N);

  // --- degrees / normalization
  fill_zero<<<(N + 255) / 256, 256, 0, stream>>>(deg, N);
  degree_kernel<<<(E + 255) / 256, 256, 0, stream>>>(ei, deg, E);
  dinv_kernel<<<(N + 255) / 256, 256, 0, stream>>>(deg, dinv, N);

  // --- GCN layer 1 (64 -> 64)
  gemm_kernel<64><<<(N + 127) / 128, 256, 0, stream>>>(bufA, W1, bufB, N);
  fill_zero<<<(int)(((long)N * 64 + 255) / 256), 256, 0, stream>>>(bufC, (long)N * 64);
  aggregate_kernel<64><<<(int)(((long)E * 16 + 255) / 256), 256, 0, stream>>>(ei, bufB, dinv, bufC, E);
  finalize_kernel<<<(int)(((long)N * 64 + 255) / 256), 256, 0, stream>>>(bufC, bufB, dinv, b1, bufA, (long)N * 64, 64);

  // --- GCN layer 2 (64 -> 32)
  float* hw2 = bufB;
  float* agg2 = bufC;
  float* h2 = bufB + (long)32 * N;
  gemm_kernel<32><<<(N + 127) / 128, 256, 0, stream>>>(bufA, W2, hw2, N);
  fill_zero<<<(int)(((long)N * 32 + 255) / 256), 256, 0, stream>>>(agg2, (long)N * 32);
  aggregate_kernel<32><<<(int)(((long)E * 8 + 255) / 256), 256, 0, stream>>>(ei, hw2, dinv, agg2, E);
  finalize_kernel<<<(int)(((long)N * 32 + 255) / 256), 256, 0, stream>>>(agg2, hw2, dinv, b2, h2, (long)N * 32, 32);

  // --- classifier + log_softmax
  classifier_kernel<<<(N + 127) / 128, 256, 0, stream>>>(h2, fcw, fcb, out_logits, N);
}